// GMBCLayer_24592982737137
// MI455X (gfx1250) — compile-verified
//
#include <hip/hip_runtime.h>

// ---------------------------------------------------------------------------
// MI455X (gfx1250) GMBC layer.
// Setup: pack lin_weight into WMMA A-fragment layout (one b64 load/fragment).
// Phase 1: grade-wise channel-mixing GEMM on v_wmma_f32_16x16x4_f32.
// Phase 2: compile-time-generated sparse Clifford bilinear (273 entries),
//          two batch rows per thread as packed f32 (V_PK_FMA_F32 path).
// Streamed x/out use non-temporal hints; VGPRs capped via waves_per_eu(4)
// (256/wave) to avoid s_set_vgpr_msb churn between WMMAs.
// ---------------------------------------------------------------------------

typedef float v2f  __attribute__((ext_vector_type(2)));
typedef float v4f  __attribute__((ext_vector_type(4)));
typedef float v8f  __attribute__((ext_vector_type(8)));
typedef float v2x  __attribute__((ext_vector_type(2)));   // packed-f32 pair

namespace ga {

constexpr int popc(int x) { int c = 0; while (x) { c += x & 1; x >>= 1; } return c; }
constexpr int swap_sign(int a, int b) {
  a >>= 1; int s = 0;
  while (a) { s += popc(a & b); a >>= 1; }
  return (s & 1) ? -1 : 1;
}

struct Tables {
  int blade[16];          // sorted blade position -> bitmask
  int idx[16];            // bitmask -> sorted position
  int grade[16];
  int ngp, njp, nnz;
  signed char ei[300], ej[300], ek[300], esign[300], etbl[300];
  short erank[300];
};

constexpr Tables make_tables() {
  Tables t{};
  int p = 0;
  for (int g = 0; g < 5; ++g)
    for (int m = 0; m < 16; ++m)
      if (popc(m) == g) { t.blade[p] = m; t.idx[m] = p; t.grade[p] = g; ++p; }

  int scomp[16] = {};
  for (int m = 0; m < 16; ++m) scomp[m] = swap_sign(m, 15 ^ m);

  bool gpp[5][5][5] = {}; bool jpp[5][5][5] = {};
  for (int j = 0; j < 16; ++j)
    for (int k = 0; k < 16; ++k) {
      const int mj = t.blade[j], mk = t.blade[k];
      if (!((mj & mk) & 1))
        gpp[t.grade[t.idx[mj ^ mk]]][t.grade[j]][t.grade[k]] = true;
      const int cj = 15 ^ mj, ck = 15 ^ mk;
      if (!(cj & ck))
        jpp[t.grade[t.idx[mj & mk]]][t.grade[j]][t.grade[k]] = true;
    }
  int gprank[5][5][5] = {}; int jprank[5][5][5] = {};
  int ng = 0, nj = 0;
  for (int a = 0; a < 5; ++a)
    for (int b = 0; b < 5; ++b)
      for (int c = 0; c < 5; ++c) {
        if (gpp[a][b][c]) gprank[a][b][c] = ng++;
        if (jpp[a][b][c]) jprank[a][b][c] = nj++;
      }
  t.ngp = ng; t.njp = nj;

  int n = 0;
  for (int j = 0; j < 16; ++j)
    for (int k = 0; k < 16; ++k) {
      const int mj = t.blade[j], mk = t.blade[k];
      if (!((mj & mk) & 1)) {
        const int i = t.idx[mj ^ mk];
        t.ei[n] = (signed char)i; t.ej[n] = (signed char)j; t.ek[n] = (signed char)k;
        t.esign[n] = (signed char)swap_sign(mj, mk);
        t.etbl[n] = 0;
        t.erank[n] = (short)gprank[t.grade[i]][t.grade[j]][t.grade[k]];
        ++n;
      }
    }
  for (int j = 0; j < 16; ++j)
    for (int k = 0; k < 16; ++k) {
      const int mj = t.blade[j], mk = t.blade[k];
      const int cj = 15 ^ mj, ck = 15 ^ mk;
      if (!(cj & ck)) {
        const int r = mj & mk;
        const int i = t.idx[r];
        t.ei[n] = (signed char)i; t.ej[n] = (signed char)j; t.ek[n] = (signed char)k;
        t.esign[n] = (signed char)(scomp[mj] * scomp[mk] * swap_sign(cj, ck) * scomp[r]);
        t.etbl[n] = 1;
        t.erank[n] = (short)jprank[t.grade[i]][t.grade[j]][t.grade[k]];
        ++n;
      }
    }
  t.nnz = n;  // 192 + 81 = 273
  return t;
}

constexpr Tables T = make_tables();
constexpr int NGP = T.ngp;
constexpr int NJP = T.njp;
constexpr int NNZ = T.nnz;

} // namespace ga

__device__ __forceinline__ int grade_of(int b) {
  return (b == 0) ? 0 : (b < 5) ? 1 : (b < 11) ? 2 : (b < 15) ? 3 : 4;
}

constexpr int CCH = 64;                 // channels
constexpr int NB  = 16;                 // batch tile per workgroup
constexpr int XROW = CCH * 16 + 4;      // padded x row stride (floats): 1028
constexpr int YOS  = 20;                // padded per-output stride in Ys
constexpr int YROW = 2 * CCH * YOS + 4; // padded y row stride: 2564
constexpr int XS_FLOATS = NB * XROW;    // 16448
constexpr int YS_FLOATS = NB * YROW;    // 41024  -> LDS ~224.5 KB < 320 KB/WGP
constexpr int NFRAG = 5 * 8 * 16;       // grades x Mtiles x ksteps = 640 fragments

// ---- setup: repack lin_weight (128,64,5) into per-lane A-fragment layout ----
// Apack[((g*8+m)*16+kk)*32 + lane] = { lw[o][k0][g], lw[o][k0+1][g] }
//   o = m*16 + (lane&15),  k0 = kk*4 + (lane>>4)*2
__global__ __launch_bounds__(256) void pack_a_kernel(
    const float* __restrict__ lw, v2f* __restrict__ ap)
{
  const int t = blockIdx.x * 256 + threadIdx.x;
  if (t >= NFRAG * 32) return;
  const int lane = t & 31;
  const int kk   = (t >> 5) & 15;
  const int m    = (t >> 9) & 7;
  const int g    = t >> 12;
  const int o    = m * 16 + (lane & 15);
  const int k0   = kk * 4 + (lane >> 4) * 2;
  v2f v;
  v.x = lw[(o * CCH + k0    ) * 5 + g];
  v.y = lw[(o * CCH + k0 + 1) * 5 + g];
  ap[t] = v;
}

__global__ __launch_bounds__(256)
__attribute__((amdgpu_waves_per_eu(4)))   // cap 1024/4 = 256 VGPRs per wave
void gmbc_kernel(
    const float* __restrict__ x,        // (B, 64, 16)
    const float* __restrict__ gpw,      // (64, NGP)
    const float* __restrict__ jpw,      // (64, NJP)
    const v2f* __restrict__ apack,      // packed A fragments (d_ws)
    float* __restrict__ out)            // (B, 64, 16)
{
  extern __shared__ float smem[];
  float* Xs = smem;               // [NB][64][16] (+pad)
  float* Ys = smem + XS_FLOATS;   // [NB][128][16] (+pad)

  const int tid  = threadIdx.x;
  const int wave = tid >> 5;
  const int lane = tid & 31;
  const long nb  = (long)blockIdx.x * NB;

  // ---- stage x tile: 16 rows x 1024 floats, coalesced non-temporal b128 ----
  {
    const v4f* xg = (const v4f*)(x + nb * (CCH * 16));
    #pragma unroll
    for (int r = 0; r < NB; ++r) {
      v4f v = __builtin_nontemporal_load(xg + r * 256 + tid);
      ((v4f*)(Xs + r * XROW))[tid] = v;
    }
  }
  __syncthreads();

  // ---- phase 1: per-blade 128x64 GEMM on WMMA f32 16x16x4 ----
  const int nrow  = lane & 15;          // N col (batch) for B/D; M row for A
  const int khalf = (lane >> 4) * 2;    // lanes 16-31 carry K = 2,3
  #pragma unroll
  for (int bb = 0; bb < 2; ++bb) {
    const int b = __builtin_amdgcn_readfirstlane(wave * 2 + bb);
    const int g = __builtin_amdgcn_readfirstlane(grade_of(b));
    const v2f* apg = apack + (size_t)g * (8 * 16 * 32) + lane; // scalar base + lane
    v8f acc[8] = {};
    #pragma unroll
    for (int kk = 0; kk < 16; ++kk) {   // K = 64 in steps of 4
      const int i0 = kk * 4 + khalf;
      v2f bf;                            // B fragment: x[n][i0..i0+1][b]
      bf.x = Xs[nrow * XROW + (i0    ) * 16 + b];
      bf.y = Xs[nrow * XROW + (i0 + 1) * 16 + b];
      #pragma unroll
      for (int m = 0; m < 8; ++m) {      // M = 128 in tiles of 16
        const v2f af = apg[(m * 16 + kk) * 32];  // one global_load_b64
        acc[m] = __builtin_amdgcn_wmma_f32_16x16x4_f32(
            false, af, false, bf, (short)0, acc[m], false, false);
      }
    }
    // scatter D (16x16 f32 layout: lanes<16 -> M=r, lanes>=16 -> M=8+r)
    const int obase = (lane >> 4) * 8;
    #pragma unroll
    for (int m = 0; m < 8; ++m) {
      #pragma unroll
      for (int r = 0; r < 8; ++r)
        Ys[nrow * YROW + (m * 16 + obase + r) * YOS + b] = acc[m][r];
    }
  }
  __syncthreads();

  // ---- phase 2: sparse Clifford bilinear, packed f32 (2 batch rows/thread) ----
  const int c = tid & 63;               // fixed channel per thread
  float wg[ga::NGP], wj[ga::NJP];
  {
    const float* gp = gpw + c * ga::NGP;
    const float* jp = jpw + c * ga::NJP;
    #pragma unroll
    for (int t = 0; t < ga::NGP; ++t) wg[t] = gp[t];
    #pragma unroll
    for (int t = 0; t < ga::NJP; ++t) wj[t] = jp[t];
  }
  #pragma unroll 1
  for (int rep = 0; rep < 2; ++rep) {
    const int n0 = (tid >> 6) + 4 * rep;      // rows n0 and n0+8 as a packed pair
    const int n1 = n0 + 8;
    const v4f* y1a = (const v4f*)(Ys + n0 * YROW + c * YOS);
    const v4f* y1b = (const v4f*)(Ys + n1 * YROW + c * YOS);
    const v4f* y2a = (const v4f*)(Ys + n0 * YROW + (CCH + c) * YOS);
    const v4f* y2b = (const v4f*)(Ys + n1 * YROW + (CCH + c) * YOS);
    v2x y1[16], y2[16], o16[16];
    #pragma unroll
    for (int q = 0; q < 4; ++q) {
      v4f a = y1a[q], bq = y1b[q];
      y1[4*q+0] = v2x{a.x, bq.x}; y1[4*q+1] = v2x{a.y, bq.y};
      y1[4*q+2] = v2x{a.z, bq.z}; y1[4*q+3] = v2x{a.w, bq.w};
      v4f d = y2a[q], e = y2b[q];
      y2[4*q+0] = v2x{d.x, e.x}; y2[4*q+1] = v2x{d.y, e.y};
      y2[4*q+2] = v2x{d.z, e.z}; y2[4*q+3] = v2x{d.w, e.w};
    }
    #pragma unroll
    for (int i = 0; i < 16; ++i) o16[i] = y2[i];     // "+ y2" residual
    #pragma unroll
    for (int t = 0; t < ga::NNZ; ++t) {
      const float w  = (ga::T.etbl[t] == 0) ? wg[ga::T.erank[t]] : wj[ga::T.erank[t]];
      const float sw = (ga::T.esign[t] < 0) ? -w : w;
      const v2x sws = v2x{sw, sw};
      o16[ga::T.ei[t]] += sws * (y1[ga::T.ej[t]] * y2[ga::T.ek[t]]);  // v_pk_fma_f32
    }
    v4f* oga = (v4f*)(out + ((nb + (long)n0) * CCH + c) * 16);
    v4f* ogb = (v4f*)(out + ((nb + (long)n1) * CCH + c) * 16);
    #pragma unroll
    for (int q = 0; q < 4; ++q) {
      v4f va, vb;
      va.x = o16[4*q+0].x; va.y = o16[4*q+1].x; va.z = o16[4*q+2].x; va.w = o16[4*q+3].x;
      vb.x = o16[4*q+0].y; vb.y = o16[4*q+1].y; vb.z = o16[4*q+2].y; vb.w = o16[4*q+3].y;
      __builtin_nontemporal_store(va, oga + q);
      __builtin_nontemporal_store(vb, ogb + q);
    }
  }
}

extern "C" void kernel_launch(void* const* d_in, const int* in_sizes, int n_in,
                              void* d_out, int out_size, void* d_ws, size_t ws_size,
                              hipStream_t stream) {
  (void)n_in; (void)out_size; (void)ws_size;
  const float* x   = (const float*)d_in[0];
  const float* gpw = (const float*)d_in[1];
  const float* jpw = (const float*)d_in[2];
  const float* lw  = (const float*)d_in[3];
  v2f* apack = (v2f*)d_ws;                         // 640*32*8 = 160 KB scratch
  float* out = (float*)d_out;

  // repack lin_weight into WMMA A-fragment layout (stream-ordered before main)
  pack_a_kernel<<<(NFRAG * 32 + 255) / 256, 256, 0, stream>>>(lw, apack);

  const int B = in_sizes[0] / (CCH * 16);          // 65536
  const int nblocks = B / NB;                      // 4096 workgroups
  const size_t shmem = (size_t)(XS_FLOATS + YS_FLOATS) * sizeof(float);

  (void)hipFuncSetAttribute((const void*)gmbc_kernel,
                            hipFuncAttributeMaxDynamicSharedMemorySize, (int)shmem);
  gmbc_kernel<<<nblocks, 256, shmem, stream>>>(x, gpw, jpw, apack, out);
}